// SparseMoeBlockHFMixtral_17867063951940
// MI455X (gfx1250) — compile-verified
//
#include <hip/hip_runtime.h>
#include <hip/hip_bf16.h>
#include <math.h>
#include <stdint.h>

#define NUM_EXPERTS 8
#define DIM 1024
#define HIDDEN 4096
#define NTOK 512
#define TILE 16
#define MAXTILES (NTOK / TILE)   /* 32 */
#define PX (DIM + 4)             /* pitch 1028: bank = 4*row+k -> conflict-free, rows 16B aligned */
#define HC 128                   /* hidden chunk per iteration */
#define PA (HC + 4)              /* pitch 132: same conflict-free geometry for act tile */

typedef __attribute__((ext_vector_type(2))) float v2f;
typedef __attribute__((ext_vector_type(8))) float v8f;

// ---------------------------------------------------------------------------
// Router: logits -> softmax top-2 -> renormalized coefs, gathered per expert.
// One block per token; wave w computes expert w's dot product.
// ---------------------------------------------------------------------------
__global__ __launch_bounds__(256) void moe_router(const float* __restrict__ x,
                                                  const float* __restrict__ gw,
                                                  int* __restrict__ counts,
                                                  int* __restrict__ tlist,
                                                  float* __restrict__ clist) {
  const int t = blockIdx.x;
  const int lane = threadIdx.x & 31;
  const int wv = threadIdx.x >> 5;  // expert index 0..7
  const float* xr = x + (size_t)t * DIM;
  const float* g = gw + (size_t)wv * DIM;
  float s = 0.f;
  for (int i = lane; i < DIM; i += 32) s += xr[i] * g[i];
#pragma unroll
  for (int off = 16; off > 0; off >>= 1) s += __shfl_xor(s, off, 32);
  __shared__ float logits[NUM_EXPERTS];
  if (lane == 0) logits[wv] = s;
  __syncthreads();
  if (threadIdx.x == 0) {
    // argmax (ties -> lowest index, matching top_k)
    int i0 = 0;
    float l0 = logits[0];
    for (int e = 1; e < NUM_EXPERTS; ++e) {
      float le = logits[e];
      if (le > l0) { l0 = le; i0 = e; }
    }
    int i1 = -1;
    float l1 = -3.4e38f;
    for (int e = 0; e < NUM_EXPERTS; ++e) {
      if (e == i0) continue;
      float le = logits[e];
      if (le > l1) { l1 = le; i1 = e; }
    }
    // softmax + top-2 renorm collapses to a 2-way logistic on (l1 - l0)
    float r = __expf(l1 - l0);
    float w0 = 1.0f / (1.0f + r);
    float w1c = r / (1.0f + r);
    int p0 = atomicAdd(&counts[i0], 1);
    tlist[i0 * NTOK + p0] = t;
    clist[i0 * NTOK + p0] = w0;
    int p1 = atomicAdd(&counts[i1], 1);
    tlist[i1 * NTOK + p1] = t;
    clist[i1 * NTOK + p1] = w1c;
  }
}

// ---------------------------------------------------------------------------
// Grouped SwiGLU FFN with V_WMMA_F32_16X16X4_F32.
// Block = (expert e, 16-token tile). 8 waves.
//   X tile staged to LDS with GLOBAL_LOAD_ASYNC_TO_LDS_B128 (ASYNCcnt path).
//   Phase A: h1/h3 chunk [16 x 128], wave w owns 16 hidden cols; K = DIM.
//   act = silu(h1)*h3 -> LDS (fast-rcp silu).
//   Phase B: out tile [16 x 1024] accumulated; wave w owns 128 D cols
//            (8 WMMA accumulators), K = hidden chunk, A reused across tiles.
// Final: scale rows by coef, atomicAdd into out (exactly 2 contribs/element).
// ---------------------------------------------------------------------------
__global__ __launch_bounds__(256) void moe_ffn(const float* __restrict__ x,
                                               const float* __restrict__ w1,
                                               const float* __restrict__ w3,
                                               const float* __restrict__ w2,
                                               const int* __restrict__ counts,
                                               const int* __restrict__ tlist,
                                               const float* __restrict__ clist,
                                               float* __restrict__ out) {
  extern __shared__ float smem[];
  float* xs = smem;                       // TILE * PX  (X tile, fp32)
  float* as_ = xs + TILE * PX;            // TILE * PA  (act chunk)
  int* toks = (int*)(as_ + TILE * PA);    // TILE
  float* cofs = (float*)(toks + TILE);    // TILE

  const int e = blockIdx.x >> 5;            // expert
  const int tile = blockIdx.x & (MAXTILES - 1);
  const int cnt = counts[e];
  if (tile * TILE >= cnt) return;

  const int tid = threadIdx.x;
  const int lane = tid & 31;
  const int wv = tid >> 5;

  if (tid < TILE) {
    int gi = tile * TILE + tid;
    if (gi < cnt) {
      toks[tid] = tlist[e * NTOK + gi];
      cofs[tid] = clist[e * NTOK + gi];
    } else {
      toks[tid] = 0;      // padded slot: valid address, zero weight
      cofs[tid] = 0.f;
    }
  }
  __syncthreads();

  // Stage X tile (16 x 1024 f32) directly into LDS: async copy, no VGPR bounce.
  for (int i = tid; i < TILE * (DIM / 4); i += 256) {
    int idx = i * 4;
    int row = idx >> 10;        // / DIM
    int col = idx & (DIM - 1);
    const float* gp = x + (size_t)toks[row] * DIM + col;
    unsigned ldsoff = (unsigned)(uintptr_t)(&xs[row * PX + col]);  // 16B aligned (PX=1028)
    asm volatile("global_load_async_to_lds_b128 %0, %1, off"
                 :: "v"(ldsoff), "v"(gp)
                 : "memory");
  }
  asm volatile("s_wait_asynccnt 0x0" ::: "memory");
  __syncthreads();

  const float* w1e = w1 + (size_t)e * HIDDEN * DIM;
  const float* w3e = w3 + (size_t)e * HIDDEN * DIM;
  const float* w2e = w2 + (size_t)e * DIM * HIDDEN;

  const int r = lane & 15;                // M (or N) index within fragment
  const int khalf = (lane >> 4) << 1;     // lanes 16-31 hold K+2..K+3

  v8f acc[8];
#pragma unroll
  for (int j = 0; j < 8; ++j) acc[j] = v8f{0.f, 0.f, 0.f, 0.f, 0.f, 0.f, 0.f, 0.f};

  for (int hb = 0; hb < HIDDEN; hb += HC) {
    // Warm L2 for the next chunk's weight rows while this chunk computes.
    if (hb + HC < HIDDEN) {
      const size_t nrow = (size_t)(hb + HC + wv * 16 + r) * DIM;
      __builtin_prefetch(w1e + nrow, 0, 1);
      __builtin_prefetch(w3e + nrow, 0, 1);
    }
    // ---- Phase A: h1 / h3 for this wave's 16 hidden columns ----
    v8f c1 = v8f{0.f, 0.f, 0.f, 0.f, 0.f, 0.f, 0.f, 0.f};
    v8f c3 = v8f{0.f, 0.f, 0.f, 0.f, 0.f, 0.f, 0.f, 0.f};
    const float* b1p = w1e + (size_t)(hb + wv * 16 + r) * DIM + khalf;
    const float* b3p = w3e + (size_t)(hb + wv * 16 + r) * DIM + khalf;
#pragma unroll 4
    for (int k = 0; k < DIM; k += 4) {
      v2f a = *(const v2f*)&xs[r * PX + k + khalf];       // A: 16x4 of X
      v2f b1 = *(const v2f*)(b1p + k);                    // B: 4x16 of W1^T
      v2f b3 = *(const v2f*)(b3p + k);
      c1 = __builtin_amdgcn_wmma_f32_16x16x4_f32(false, a, false, b1, (short)0, c1, false, false);
      c3 = __builtin_amdgcn_wmma_f32_16x16x4_f32(false, a, false, b3, (short)0, c3, false, false);
    }
    __syncthreads();  // prior phase-B reads of as_ are done
#pragma unroll
    for (int v = 0; v < 8; ++v) {
      float h1v = c1[v];
      float h3v = c3[v];
      // silu via fast reciprocal (v_rcp_f32, ~1 ulp) instead of IEEE divide
      float sg = h1v * __builtin_amdgcn_rcpf(1.0f + __expf(-h1v));
      int m = v + ((lane >> 4) << 3);                     // C layout: M = vgpr + 8*(lane/16)
      int n = lane & 15;
      as_[m * PA + wv * 16 + n] = sg * h3v;
    }
    __syncthreads();

    // ---- Phase B: out += act_chunk @ W2^T (A reused across 8 N-tiles) ----
#pragma unroll 2
    for (int kk = 0; kk < HC; kk += 4) {
      v2f a = *(const v2f*)&as_[r * PA + kk + khalf];
      const float* wb = w2e + (size_t)(wv * 128 + r) * HIDDEN + hb + kk + khalf;
#pragma unroll
      for (int j = 0; j < 8; ++j) {
        v2f b = *(const v2f*)(wb + (size_t)j * 16 * HIDDEN);
        acc[j] = __builtin_amdgcn_wmma_f32_16x16x4_f32(false, a, false, b, (short)0, acc[j], false, false);
      }
    }
  }
  __syncthreads();

  // ---- Writeback: scale by routing coef, accumulate into global output ----
#pragma unroll
  for (int j = 0; j < 8; ++j) {
    int nd = wv * 128 + j * 16 + (lane & 15);
#pragma unroll
    for (int v = 0; v < 8; ++v) {
      int m = v + ((lane >> 4) << 3);
      float val = acc[j][v] * cofs[m];
      atomicAdd(out + (size_t)toks[m] * DIM + nd, val);
    }
  }
}

// ---------------------------------------------------------------------------
extern "C" void kernel_launch(void* const* d_in, const int* in_sizes, int n_in,
                              void* d_out, int out_size, void* d_ws, size_t ws_size,
                              hipStream_t stream) {
  const float* x  = (const float*)d_in[0];   // [512, 1024]
  const float* gw = (const float*)d_in[1];   // [8, 1024]
  const float* w1 = (const float*)d_in[2];   // [8, 4096, 1024]
  const float* w3 = (const float*)d_in[3];   // [8, 4096, 1024]
  const float* w2 = (const float*)d_in[4];   // [8, 1024, 4096]
  float* out = (float*)d_out;                // [512, 1024]

  int* counts = (int*)d_ws;                            // 8 ints
  int* tlist = counts + NUM_EXPERTS;                   // 8*512 ints
  float* clist = (float*)(tlist + NUM_EXPERTS * NTOK); // 8*512 floats

  hipMemsetAsync(counts, 0, NUM_EXPERTS * sizeof(int), stream);
  hipMemsetAsync(out, 0, (size_t)out_size * sizeof(float), stream);

  moe_router<<<NTOK, 256, 0, stream>>>(x, gw, counts, tlist, clist);

  size_t smem = (size_t)(TILE * PX + TILE * PA + TILE + TILE) * sizeof(float);
  moe_ffn<<<NUM_EXPERTS * MAXTILES, 256, smem, stream>>>(x, w1, w3, w2, counts,
                                                         tlist, clist, out);
}